// MyModel_61933428413171
// MI455X (gfx1250) — compile-verified
//
#include <hip/hip_runtime.h>
#include <hip/hip_bf16.h>
#include <math.h>

typedef __attribute__((ext_vector_type(2))) float v2f;
typedef __attribute__((ext_vector_type(4))) float v4f;
typedef __attribute__((ext_vector_type(8))) float v8f;

#define TPB 256
#define WAVES_PER_BLOCK (TPB / 32)
#define MAX_BLOCKS 512
#define TILE 256  // elements per wave per iteration (2 x b128 per lane)

// ||t||^2 via V_WMMA_F32_16X16X4_F32 with B == ones(4,16):
// D[i][j] = sum_k A[i][k], so column 0 (lane 0 -> M=0..7, lane 16 -> M=8..15)
// of the f32 accumulator carries the running total. Each lane feeds the
// pair-reduced squares of 8 elements (two b128 loads) into its 2 A-VGPRs,
// so one WMMA consumes 256 elements per wave.
__global__ __launch_bounds__(TPB) void tnorm2_wmma_kernel(
    const float* __restrict__ t, int n, float* __restrict__ partials) {
  const int lane  = threadIdx.x & 31;
  const int wave  = threadIdx.x >> 5;
  const long long gwave  = (long long)blockIdx.x * WAVES_PER_BLOCK + wave;
  const long long nwaves = (long long)gridDim.x * WAVES_PER_BLOCK;

  v8f c = {0.f, 0.f, 0.f, 0.f, 0.f, 0.f, 0.f, 0.f};
  const v2f ones = {1.0f, 1.0f};

  const long long full = (long long)n & ~(long long)(TILE - 1);

  // Wave-uniform bound; base is a multiple of TILE and base < full implies
  // base + TILE <= full <= n, so both b128 loads are unconditionally in
  // bounds: no divergence, EXEC stays all-1s for the WMMA.
  for (long long base = gwave * TILE; base < full; base += nwaves * TILE) {
    const float* p = t + base + 4 * lane;          // 16B aligned
    v4f u = *(const v4f*)(p);                      // elements [0,128)
    v4f w = *(const v4f*)(p + 128);                // elements [128,256)
    v2f a;
    a.x = fmaf(u.x, u.x, fmaf(u.y, u.y, fmaf(w.x, w.x, w.y * w.y)));
    a.y = fmaf(u.z, u.z, fmaf(u.w, u.w, fmaf(w.z, w.z, w.w * w.w)));
    // 8 args: (neg_a, A, neg_b, B, c_mod, C, reuse_a, reuse_b)
    c = __builtin_amdgcn_wmma_f32_16x16x4_f32(
        false, a, false, ones, (short)0, c, false, false);
  }

  // Tail (< TILE elements): handled once, by global wave 0, guarded scalar
  // loads outside the WMMA region.
  float tailacc = 0.f;
  if (gwave == 0) {
    for (long long i = full + lane; i < n; i += 32) {
      float x = t[i];
      tailacc = fmaf(x, x, tailacc);
    }
  }
  #pragma unroll
  for (int off = 16; off > 0; off >>= 1)
    tailacc += __shfl_down(tailacc, off, 32);      // total on lane 0

  // Column 0 of D: lane 0 holds M=0..7, lane 16 holds M=8..15.
  float rowsum = c.s0 + c.s1 + c.s2 + c.s3 + c.s4 + c.s5 + c.s6 + c.s7;
  float colsum = rowsum + __shfl(rowsum, lane ^ 16, 32);

  __shared__ float wsum[WAVES_PER_BLOCK];
  if (lane == 0) wsum[wave] = colsum + tailacc;
  __syncthreads();

  if (threadIdx.x == 0) {
    float s = 0.f;
    for (int w = 0; w < WAVES_PER_BLOCK; ++w) s += wsum[w];
    partials[blockIdx.x] = s;  // every block writes its slot every call
  }
}

// Deterministic fixed-order finish: sum block partials, fold in ||A||_F^2,
// write sqrt(||t||^2 * ||A||_F^2).
__global__ void tnorm2_finalize_kernel(const float* __restrict__ partials,
                                       int nblocks,
                                       const float* __restrict__ A, int alen,
                                       float* __restrict__ out) {
  if (threadIdx.x == 0 && blockIdx.x == 0) {
    float tsum = 0.f;
    for (int i = 0; i < nblocks; ++i) tsum += partials[i];
    float asum = 0.f;
    for (int i = 0; i < alen; ++i) {
      float v = A[i];
      asum = fmaf(v, v, asum);
    }
    out[0] = sqrtf(tsum * asum);
  }
}

extern "C" void kernel_launch(void* const* d_in, const int* in_sizes, int n_in,
                              void* d_out, int out_size, void* d_ws,
                              size_t ws_size, hipStream_t stream) {
  const float* t = (const float*)d_in[0];  // [N] float32
  const float* A = (const float*)d_in[1];  // [5,5] float32
  const int n    = in_sizes[0];
  const int alen = in_sizes[1];

  int blocks = MAX_BLOCKS;
  size_t max_by_ws = ws_size / sizeof(float);
  if ((size_t)blocks > max_by_ws) blocks = (int)max_by_ws;
  if (blocks < 1) blocks = 1;

  float* partials = (float*)d_ws;

  tnorm2_wmma_kernel<<<blocks, TPB, 0, stream>>>(t, n, partials);
  tnorm2_finalize_kernel<<<1, 32, 0, stream>>>(partials, blocks, A, alen,
                                               (float*)d_out);
}